// AUCLoss_89275190214933
// MI455X (gfx1250) — compile-verified
//
#include <hip/hip_runtime.h>
#include <hip/hip_bf16.h>
#include <math.h>
#include <stdint.h>

#define GAMMA 0.15f
#define ALPHA 0.6f

typedef __attribute__((ext_vector_type(2))) float v2f;
typedef __attribute__((ext_vector_type(8))) float v8f;

// 16-byte int vector type matching the async-to-LDS builtin's parameter type
typedef int v4i_ __attribute__((vector_size(16)));
typedef __attribute__((address_space(1))) v4i_ as1_v4i;
typedef __attribute__((address_space(3))) v4i_ as3_v4i;

// ---------------------------------------------------------------------------
// Kernel 1: per-element prep.
//   pred  = sigmoid(y_pred[:,1] - y_pred[:,0])
//   t[n]  = (y==1) ? pred + GAMMA : -1e30   (row operand, "neg" rows)
//   x[n]  = (y==0) ? pred         : +1e30   (col operand, "pos" cols)
//   bce_part[block] = block sum of (y*log(p) + (1-y)*log1p(-p)), clamped
// Padded tail [N, Npad) gets sentinels so the WMMA sweep needs no bounds mask.
// ---------------------------------------------------------------------------
__global__ void auc_prep(const float* __restrict__ yp,
                         const int*   __restrict__ yt,
                         float* __restrict__ t,
                         float* __restrict__ x,
                         float* __restrict__ bce_part,
                         int N, int Npad) {
    int n = blockIdx.x * blockDim.x + threadIdx.x;
    float bce = 0.0f;
    if (n < Npad) {
        if (n < N) {
            float a = yp[2 * n + 0];
            float b = yp[2 * n + 1];
            float z = b - a;                       // logit of class 1
            float p = 1.0f / (1.0f + expf(-z));    // softmax(...)[:,1]
            int   y = yt[n];
            float logp   = fmaxf(logf(p),     -100.0f);
            float log1mp = fmaxf(log1pf(-p),  -100.0f);
            bce  = (y == 1) ? logp : log1mp;       // y in {0,1}
            t[n] = (y == 1) ? (p + GAMMA) : -1.0e30f;
            x[n] = (y == 0) ? p           :  1.0e30f;
        } else {
            t[n] = -1.0e30f;
            x[n] =  1.0e30f;
        }
    }
    __shared__ float sm[256];
    sm[threadIdx.x] = bce;
    __syncthreads();
    for (int s = 128; s > 0; s >>= 1) {
        if ((int)threadIdx.x < s) sm[threadIdx.x] += sm[threadIdx.x + s];
        __syncthreads();
    }
    if (threadIdx.x == 0) bce_part[blockIdx.x] = sm[0];
}

// ---------------------------------------------------------------------------
// Kernel 2: pairwise sweep. One wave per 16-row strip; loop over all 16-col
// tiles. D[i,j] = t_i - x_j via V_WMMA_F32_16X16X4_F32 with A=[t,1], B=[1;-x].
// Sentinels force d<0 for every invalid pair. x[] is staged once per block
// into LDS with global_load_async_to_lds_b128 (ASYNCcnt). Count accumulates
// via wave ballot -> SGPR (co-issues with VALU).
// ---------------------------------------------------------------------------
__global__ void __launch_bounds__(256)
auc_pair(const float* __restrict__ t,
         const float* __restrict__ x,
         float*    __restrict__ psum,
         unsigned* __restrict__ pcnt,
         int strips, int Npad) {
    extern __shared__ float xs[];

    // ---- stage x[0..Npad) into LDS, all 256 threads cooperate ----
#if __has_builtin(__builtin_amdgcn_global_load_async_to_lds_b128)
    for (int i = (int)threadIdx.x * 4; i < Npad; i += 1024) {
        __builtin_amdgcn_global_load_async_to_lds_b128(
            (as1_v4i*)(uintptr_t)(x + i),
            (as3_v4i*)(uintptr_t)(xs + i),
            /*offset=*/0, /*cpol=*/0);
    }
  #if __has_builtin(__builtin_amdgcn_s_wait_asynccnt)
    __builtin_amdgcn_s_wait_asynccnt(0);
  #else
    asm volatile("s_wait_asynccnt 0x0" ::: "memory");
  #endif
#else
    for (int i = (int)threadIdx.x * 4; i < Npad; i += 1024) {
        float4 v = *(const float4*)(x + i);
        *(float4*)(xs + i) = v;
    }
#endif
    __syncthreads();   // all waves reach this before any early exit

    int wid  = (int)(blockIdx.x * blockDim.x + threadIdx.x) >> 5;
    int lane = (int)(threadIdx.x & 31u);
    if (wid >= strips) return;                 // wave-uniform exit (after barrier)

    int  lm = lane & 15;
    bool lo = lane < 16;

    // A operand: lanes 0-15 hold K=0 (t_i) / K=1 (1.0); lanes 16-31 -> K=2,3 = 0
    float tv = t[wid * 16 + lm];
    v2f A;
    A.x = lo ? tv   : 0.0f;
    A.y = lo ? 1.0f : 0.0f;

    v8f C = {};
    float    sum = 0.0f;
    unsigned cnt = 0u;                          // wave-uniform (ballot-based)

    for (int tile = 0; tile < strips; ++tile) {
        float xv = xs[tile * 16 + lm];          // ds_load from staged LDS copy

        // B operand: lanes 0-15 hold K=0 (1.0) / K=1 (-x_j); lanes 16-31 -> 0
        v2f B;
        B.x = lo ? 1.0f : 0.0f;
        B.y = lo ? -xv  : 0.0f;

        // D[i,j] = t_i - x_j, full 16x16 tile in 8 VGPRs across the wave
        v8f d = __builtin_amdgcn_wmma_f32_16x16x4_f32(
            false, A, false, B, (short)0, C, false, false);

#pragma unroll
        for (int k = 0; k < 8; ++k) {
            float dk = d[k];
            bool  g  = dk > 0.0f;               // one v_cmp feeds ballot + select
#if __has_builtin(__builtin_amdgcn_ballot_w32)
            unsigned m = __builtin_amdgcn_ballot_w32(g);
#else
            unsigned m = (unsigned)__ballot(g);
#endif
            cnt += (unsigned)__builtin_popcount(m);   // SGPR-side, co-issues
            float dd = g ? dk : 0.0f;                 // v_cndmask off same cmp
            sum = fmaf(dd, dd, sum);
        }
    }

    // wave32 butterfly reduction for sum (cnt is already wave-uniform)
    for (int off = 16; off >= 1; off >>= 1)
        sum += __shfl_xor(sum, off, 32);
    if (lane == 0) {
        psum[wid] = sum;
        pcnt[wid] = cnt;
    }
}

// ---------------------------------------------------------------------------
// Kernel 3: final reduction of partials + scalar loss.
// ---------------------------------------------------------------------------
__global__ void auc_final(const float* __restrict__ bce_part, int nb,
                          const float* __restrict__ psum,
                          const unsigned* __restrict__ pcnt, int nw,
                          float* __restrict__ out, int N) {
    __shared__ float    sb[256];
    __shared__ float    ss[256];
    __shared__ unsigned sc[256];
    int tid = threadIdx.x;
    float b = 0.0f, s = 0.0f;
    unsigned c = 0u;
    for (int i = tid; i < nb; i += 256) b += bce_part[i];
    for (int i = tid; i < nw; i += 256) { s += psum[i]; c += pcnt[i]; }
    sb[tid] = b; ss[tid] = s; sc[tid] = c;
    __syncthreads();
    for (int st = 128; st > 0; st >>= 1) {
        if (tid < st) {
            sb[tid] += sb[tid + st];
            ss[tid] += ss[tid + st];
            sc[tid] += sc[tid + st];
        }
        __syncthreads();
    }
    if (tid == 0) {
        float bce = -sb[0] / (float)N;
        unsigned count = sc[0];
        float denom = (float)(count > 0u ? count : 1u);
        float auc = ss[0] / denom;
        out[0] = (count > 0u) ? (ALPHA * bce + (1.0f - ALPHA) * auc) : bce;
    }
}

// ---------------------------------------------------------------------------
extern "C" void kernel_launch(void* const* d_in, const int* in_sizes, int n_in,
                              void* d_out, int out_size, void* d_ws, size_t ws_size,
                              hipStream_t stream) {
    const float* yp = (const float*)d_in[0];   // (N,2) float32
    const int*   yt = (const int*)d_in[1];     // (N,)  integer

    int N      = in_sizes[1];
    int strips = (N + 15) / 16;
    int Npad   = strips * 16;
    int B1     = (Npad + 255) / 256;

    // ws layout (floats): t[Npad] | x[Npad] | bce_part[B1] | psum[strips] | pcnt[strips]
    float*    t        = (float*)d_ws;
    float*    x        = t + Npad;
    float*    bce_part = x + Npad;
    float*    psum     = bce_part + B1;
    unsigned* pcnt     = (unsigned*)(psum + strips);

    auc_prep<<<B1, 256, 0, stream>>>(yp, yt, t, x, bce_part, N, Npad);

    int B2 = (strips + 7) / 8;                 // 8 waves (256 threads) per block
    auc_pair<<<B2, 256, (size_t)Npad * sizeof(float), stream>>>(
        t, x, psum, pcnt, strips, Npad);

    auc_final<<<1, 256, 0, stream>>>(bce_part, B1, psum, pcnt, strips,
                                     (float*)d_out, N);
}